// MLSLoss_29480655520004
// MI455X (gfx1250) — compile-verified
//
#include <hip/hip_runtime.h>
#include <hip/hip_bf16.h>
#include <math.h>

#define N_ROWS 768
#define DIM 512
#define BLK 256
#define WAVES (BLK / 32)

typedef float v2f  __attribute__((ext_vector_type(2)));
typedef float v8f  __attribute__((ext_vector_type(8)));
typedef unsigned int u32x4 __attribute__((ext_vector_type(4)));
typedef int   i32x4 __attribute__((ext_vector_type(4)));
typedef int   i32x8 __attribute__((ext_vector_type(8)));

#ifndef __has_builtin
#define __has_builtin(x) 0
#endif

// ---------------------------------------------------------------------------
// Kernel 1: row-wise L2 normalize mu_X -> mu, and sig = exp(log_sigma_sq)
// ---------------------------------------------------------------------------
__global__ void mls_prep_kernel(const float* __restrict__ muX,
                                const float* __restrict__ lss,
                                float* __restrict__ mu,
                                float* __restrict__ sig) {
    __shared__ float red[BLK];
    __shared__ float s_inv;
    const int i = blockIdx.x;
    const int t = threadIdx.x;
    const size_t base = (size_t)i * DIM;

    const float x0 = muX[base + t];
    const float x1 = muX[base + t + BLK];
    red[t] = x0 * x0 + x1 * x1;
    __syncthreads();
    for (int s = BLK / 2; s > 0; s >>= 1) {
        if (t < s) red[t] += red[t + s];
        __syncthreads();
    }
    if (t == 0) {
        const float nrm = sqrtf(red[0]);
        s_inv = 1.0f / fmaxf(nrm, 1e-12f);
    }
    __syncthreads();
    const float inv = s_inv;
    mu[base + t]        = x0 * inv;
    mu[base + t + BLK]  = x1 * inv;
    sig[base + t]       = expf(lss[base + t]);
    sig[base + t + BLK] = expf(lss[base + t + BLK]);
}

// ---------------------------------------------------------------------------
// TDM: stage one 512-float row into LDS via tensor_load_to_lds (D# per ISA
// CDNA5 ISA; 1D tile, data_size=4B, no cluster multicast).
// 6-arg builtin form (clang-23 / therock-10.0 headers).
// ---------------------------------------------------------------------------
#if __has_builtin(__builtin_amdgcn_tensor_load_to_lds)
__device__ __forceinline__ void tdm_load_row(unsigned lds_off,
                                             const float* __restrict__ gptr,
                                             int nelem) {
    const unsigned long long ga = (unsigned long long)(uintptr_t)gptr;
    u32x4 g0;
    g0[0] = 1u;                                           // count=1 (valid D#)
    g0[1] = lds_off;                                      // lds_addr (bytes)
    g0[2] = (unsigned)(ga & 0xffffffffu);                 // global_addr[31:0]
    g0[3] = (unsigned)((ga >> 32) & 0x1ffffffu) | (2u << 30); // addr[56:32], type=2
    i32x8 g1;
    g1[0] = (int)(2u << 16);                              // data_size=2 (4 bytes)
    g1[1] = (int)(((unsigned)nelem & 0xffffu) << 16);     // tensor_dim0[15:0]
    g1[2] = (int)((((unsigned)nelem >> 16) & 0xffffu) | (1u << 16)); // dim0 hi | tensor_dim1=1
    g1[3] = (int)(((unsigned)nelem & 0xffffu) << 16);     // tile_dim0 = nelem
    g1[4] = 1;                                            // tile_dim1=1, tile_dim2=0
    g1[5] = nelem;                                        // tensor_dim0_stride[31:0]
    g1[6] = 0;                                            // stride hi bits
    g1[7] = 0;
    i32x4 gz4 = {0, 0, 0, 0};
    i32x8 gz8 = {0, 0, 0, 0, 0, 0, 0, 0};
    __builtin_amdgcn_tensor_load_to_lds(g0, g1, gz4, gz4, gz8, 0);
}
#endif

// ---------------------------------------------------------------------------
// Kernel 2: one block per row i. Waves scan candidate j; masked pairs only.
// Per-lane partial sums; block reduction done on the matrix pipe via
// v_wmma_f32_16x16x4_f32 with A = ones (column-sum trick), exact in f32.
// ---------------------------------------------------------------------------
__global__ void mls_pairs_kernel(const float* __restrict__ mu,
                                 const float* __restrict__ sig,
                                 const int* __restrict__ gty,
                                 float* __restrict__ bsum,
                                 int* __restrict__ bcnt) {
    __shared__ float s_mu[DIM];
    __shared__ float s_sig[DIM];
    __shared__ float s_part[BLK];
    __shared__ int   s_cnt[WAVES];

    const int i    = blockIdx.x;
    const int tid  = threadIdx.x;
    const int wave = tid >> 5;
    const int lane = tid & 31;

#if __has_builtin(__builtin_amdgcn_tensor_load_to_lds)
    if (tid < 32) {   // single wave issues both TDM descriptors
        tdm_load_row((unsigned)(uintptr_t)(void*)s_mu,  mu  + (size_t)i * DIM, DIM);
        tdm_load_row((unsigned)(uintptr_t)(void*)s_sig, sig + (size_t)i * DIM, DIM);
        __builtin_amdgcn_s_wait_tensorcnt(0);
    }
#else
    for (int d = tid; d < DIM; d += BLK) {
        s_mu[d]  = mu[(size_t)i * DIM + d];
        s_sig[d] = sig[(size_t)i * DIM + d];
    }
#endif
    __syncthreads();

    const int gi = gty[i];
    float acc = 0.0f;
    int cnt = 0;

    for (int j = wave; j < N_ROWS; j += WAVES) {   // j uniform per wave
        if (j == i || gty[j] != gi) continue;
        cnt++;
        const float* __restrict__ muj = mu  + (size_t)j * DIM;
        const float* __restrict__ sgj = sig + (size_t)j * DIM;
#pragma unroll
        for (int t = 0; t < DIM / 32; ++t) {
            const int d = lane + (t << 5);
            const float md = s_mu[d] - muj[d];
            const float ss = s_sig[d] + sgj[d];
            acc += md * md / (1e-10f + ss) + logf(ss);
        }
    }

    s_part[tid] = acc;
    if (lane == 0) s_cnt[wave] = cnt;
    __syncthreads();

    if (tid < 32) {   // wave 0, EXEC all ones (WMMA requirement)
        float total;
#if __has_builtin(__builtin_amdgcn_wmma_f32_16x16x4_f32)
        v8f c = {};
        v2f a; a[0] = 1.0f; a[1] = 1.0f;             // A = ones(16x4)
#pragma unroll
        for (int ch = 0; ch < 4; ++ch) {             // 4 WMMAs fold 256 values
            v2f b;
            b[0] = s_part[ch * 64 + tid];
            b[1] = s_part[ch * 64 + 32 + tid];
            c = __builtin_amdgcn_wmma_f32_16x16x4_f32(
                    false, a, false, b, (short)0, c, false, false);
        }
        // c[0]: lanes 0..15 = colsums, lanes 16..31 duplicate them -> /2
        float v = c[0];
        for (int off = 16; off > 0; off >>= 1) v += __shfl_xor(v, off, 32);
        total = 0.5f * v;
#else
        float v = 0.0f;
        for (int k = 0; k < BLK / 32; ++k) v += s_part[tid + 32 * k];
        for (int off = 16; off > 0; off >>= 1) v += __shfl_xor(v, off, 32);
        total = v;
#endif
        if (tid == 0) {
            bsum[i] = total;
            int ctot = 0;
            for (int w = 0; w < WAVES; ++w) ctot += s_cnt[w];
            bcnt[i] = ctot;
        }
    }
}

// ---------------------------------------------------------------------------
// Kernel 3: fixed-order final reduction -> deterministic scalar output
// ---------------------------------------------------------------------------
__global__ void mls_finalize_kernel(const float* __restrict__ bsum,
                                    const int* __restrict__ bcnt,
                                    float* __restrict__ out) {
    if (threadIdx.x == 0 && blockIdx.x == 0) {
        float s = 0.0f;
        int   c = 0;
        for (int k = 0; k < N_ROWS; ++k) { s += bsum[k]; c += bcnt[k]; }
        out[0] = s / (float)c;
    }
}

extern "C" void kernel_launch(void* const* d_in, const int* in_sizes, int n_in,
                              void* d_out, int out_size, void* d_ws, size_t ws_size,
                              hipStream_t stream) {
    const float* muX = (const float*)d_in[0];
    const float* lss = (const float*)d_in[1];
    const int*   gty = (const int*)d_in[2];
    float* out = (float*)d_out;

    float* mu   = (float*)d_ws;
    float* sig  = mu  + (size_t)N_ROWS * DIM;
    float* bsum = sig + (size_t)N_ROWS * DIM;
    int*   bcnt = (int*)(bsum + N_ROWS);

    mls_prep_kernel<<<N_ROWS, BLK, 0, stream>>>(muX, lss, mu, sig);
    mls_pairs_kernel<<<N_ROWS, BLK, 0, stream>>>(mu, sig, gty, bsum, bcnt);
    mls_finalize_kernel<<<1, 32, 0, stream>>>(bsum, bcnt, out);
}